// KnnModuleSingle_14053132992611
// MI455X (gfx1250) — compile-verified
//
#include <hip/hip_runtime.h>
#include <hip/hip_bf16.h>
#include <float.h>

typedef __attribute__((ext_vector_type(2))) float v2f;
typedef __attribute__((ext_vector_type(8))) float v8f;

#define B_Q     2048
#define DIM     1024
#define NTRAIN  100000
#define NCLS    1000
#define TOPK    201
#define CHUNK   6272          // 128 * 49 columns per chunk
#define NCHUNK  16            // 16 * 6272 = 100352 >= 100000
#define SEG     4096
#define HSTRIDE 208
#define NEG_INF (-3.402823466e38f)
#define INV_T   (1.0f / 0.07f)

// ---------------------------------------------------------------------------
// GEMM chunk: sims[m, n] = dot(features[m,:], train[chunk_off+n,:])
// Exact f32 via V_WMMA_F32_16X16X4_F32. Each wave computes a 32x32 tile
// (4 accumulators); per K-step: 2 A-frag + 2 B-frag loads feed 4 WMMAs
// (1 load/WMMA, 2x register reuse per fragment).
// A frag (16x4 f32): lanes 0-15 -> M=lane, K={k,k+1}; lanes 16-31 -> M=lane-16, K={k+2,k+3}
// B frag (4x16 f32): lanes 0-15 -> N=lane, K={k,k+1}; lanes 16-31 -> N=lane-16, K={k+2,k+3}
// C/D (16x16 f32):   VGPR r: lanes 0-15 -> M=r, N=lane; lanes 16-31 -> M=r+8, N=lane-16
// ---------------------------------------------------------------------------
__global__ __launch_bounds__(256) void gemm_chunk_kernel(
    const float* __restrict__ feat, const float* __restrict__ train,
    float* __restrict__ sims, int chunk_off)
{
  const int lane = threadIdx.x & 31;
  const int wave = threadIdx.x >> 5;
  const int half = lane >> 4;      // 0 or 1
  const int l16  = lane & 15;
  const int wm   = wave >> 2;      // 0..1  (M sub-tile)
  const int wn   = wave & 3;       // 0..3  (N sub-tile)
  const int mw   = (blockIdx.y << 6) + (wm << 5);   // 64 rows per block, 32 per wave
  const int nw   = (blockIdx.x << 7) + (wn << 5);   // 128 cols per block, 32 per wave
  const int t0   = chunk_off + nw + l16;            // global train row, B frag 0
  const int t1   = t0 + 16;                         // global train row, B frag 1

  const float* __restrict__ a0p = feat + (size_t)(mw + l16) * DIM + 2 * half;
  const float* __restrict__ a1p = a0p + (size_t)16 * DIM;
  const float* __restrict__ b0p = train + (size_t)(t0 < NTRAIN ? t0 : 0) * DIM + 2 * half;
  const float* __restrict__ b1p = train + (size_t)(t1 < NTRAIN ? t1 : 0) * DIM + 2 * half;

  v8f acc00 = {}, acc01 = {}, acc10 = {}, acc11 = {};
#pragma unroll 4
  for (int k = 0; k < DIM; k += 4) {
    v2f a0 = *(const v2f*)(a0p + k);
    v2f a1 = *(const v2f*)(a1p + k);
    v2f b0 = *(const v2f*)(b0p + k);
    v2f b1 = *(const v2f*)(b1p + k);
    acc00 = __builtin_amdgcn_wmma_f32_16x16x4_f32(false, a0, false, b0, (short)0, acc00, false, false);
    acc01 = __builtin_amdgcn_wmma_f32_16x16x4_f32(false, a0, false, b1, (short)0, acc01, false, false);
    acc10 = __builtin_amdgcn_wmma_f32_16x16x4_f32(false, a1, false, b0, (short)0, acc10, false, false);
    acc11 = __builtin_amdgcn_wmma_f32_16x16x4_f32(false, a1, false, b1, (short)0, acc11, false, false);
  }

  const bool ok0 = t0 < NTRAIN;
  const bool ok1 = t1 < NTRAIN;
#pragma unroll
  for (int r = 0; r < 8; ++r) {
    int m0 = mw + r + (half << 3);
    int m1 = m0 + 16;
    int c0 = nw + l16;
    int c1 = c0 + 16;
    sims[(size_t)m0 * CHUNK + c0] = ok0 ? acc00[r] : NEG_INF;
    sims[(size_t)m0 * CHUNK + c1] = ok1 ? acc01[r] : NEG_INF;
    sims[(size_t)m1 * CHUNK + c0] = ok0 ? acc10[r] : NEG_INF;
    sims[(size_t)m1 * CHUNK + c1] = ok1 ? acc11[r] : NEG_INF;
  }
}

// ---------------------------------------------------------------------------
// Streaming top-201 merge: one block per query row.
// Candidates above the running heap-min are compacted into LDS, then merged
// serially into a 201-entry min-heap (root = current threshold).
// ---------------------------------------------------------------------------
__global__ __launch_bounds__(256) void topk_update_kernel(
    const float* __restrict__ sims, float* __restrict__ hv_g,
    int* __restrict__ hi_g, int* __restrict__ hs_g, int chunk, int chunk_off)
{
  __shared__ float hval[TOPK];
  __shared__ int   hidx[TOPK];
  __shared__ int   hsize;
  __shared__ float cval[SEG];
  __shared__ int   cidx[SEG];
  __shared__ int   ccount;

  const int row = blockIdx.x;
  const int tid = threadIdx.x;

  if (chunk == 0) {
    if (tid == 0) hsize = 0;
  } else {
    for (int i = tid; i < TOPK; i += 256) {
      hval[i] = hv_g[(size_t)row * HSTRIDE + i];
      hidx[i] = hi_g[(size_t)row * HSTRIDE + i];
    }
    if (tid == 0) hsize = hs_g[row];
  }

  const float* __restrict__ srow = sims + (size_t)row * CHUNK;
  int limit = NTRAIN - chunk_off; if (limit > CHUNK) limit = CHUNK;

  for (int seg = 0; seg < limit; seg += SEG) {
    __syncthreads();                       // heap state visible
    float thr = (hsize >= TOPK) ? hval[0] : NEG_INF;
    if (tid == 0) ccount = 0;
    __syncthreads();
    int segEnd = seg + SEG; if (segEnd > limit) segEnd = limit;
    for (int j = seg + tid; j < segEnd; j += 256) {
      float v = srow[j];
      if (v > thr) {
        int p = atomicAdd(&ccount, 1);     // p < SEG by construction
        cval[p] = v; cidx[p] = chunk_off + j;
      }
    }
    __syncthreads();
    if (tid == 0) {
      int n = ccount;
      int hs = hsize;
      for (int c = 0; c < n; ++c) {
        float v = cval[c]; int ix = cidx[c];
        if (hs < TOPK) {                   // append + sift up
          int i = hs++;
          hval[i] = v; hidx[i] = ix;
          while (i > 0) {
            int p = (i - 1) >> 1;
            if (hval[p] > hval[i]) {
              float tv = hval[p]; hval[p] = hval[i]; hval[i] = tv;
              int   ti = hidx[p]; hidx[p] = hidx[i]; hidx[i] = ti;
              i = p;
            } else break;
          }
        } else if (v > hval[0]) {          // replace root + sift down
          hval[0] = v; hidx[0] = ix;
          int i = 0;
          for (;;) {
            int l = 2 * i + 1, r2 = 2 * i + 2, s = i;
            if (l  < TOPK && hval[l]  < hval[s]) s = l;
            if (r2 < TOPK && hval[r2] < hval[s]) s = r2;
            if (s == i) break;
            float tv = hval[s]; hval[s] = hval[i]; hval[i] = tv;
            int   ti = hidx[s]; hidx[s] = hidx[i]; hidx[i] = ti;
            i = s;
          }
        }
      }
      hsize = hs;
    }
  }
  __syncthreads();
  for (int i = tid; i < TOPK; i += 256) {
    hv_g[(size_t)row * HSTRIDE + i] = hval[i];
    hi_g[(size_t)row * HSTRIDE + i] = hidx[i];
  }
  if (tid == 0) hs_g[row] = hsize;
}

// ---------------------------------------------------------------------------
// Finalize: bitonic-sort 201 survivors (desc value, asc index on ties),
// drop rank 0, softmax over ranks 1..200, deterministic vote accumulation.
// ---------------------------------------------------------------------------
__global__ __launch_bounds__(256) void finalize_kernel(
    const float* __restrict__ hv_g, const int* __restrict__ hi_g,
    const int* __restrict__ labels, float* __restrict__ out)
{
  __shared__ float sv[256];
  __shared__ int   si[256];
  __shared__ float red[256];
  __shared__ float wsh[256];
  __shared__ int   lbl[256];
  __shared__ float votes[4 * NCLS];

  const int row = blockIdx.x;
  const int t   = threadIdx.x;

  if (t < TOPK) {
    sv[t] = hv_g[(size_t)row * HSTRIDE + t];
    si[t] = hi_g[(size_t)row * HSTRIDE + t];
  } else {
    sv[t] = NEG_INF; si[t] = 0x7FFFFFFF;
  }
  __syncthreads();

  // bitonic sort, 256 elements, descending by value (index tie-break)
  for (int k = 2; k <= 256; k <<= 1) {
    for (int j = k >> 1; j > 0; j >>= 1) {
      int ixj = t ^ j;
      if (ixj > t) {
        float va = sv[t], vb = sv[ixj];
        int   ia = si[t], ib = si[ixj];
        bool aFirst = (va > vb) || (va == vb && ia < ib);
        bool bFirst = (vb > va) || (vb == va && ib < ia);
        bool asc = ((t & k) == 0);
        if (asc ? bFirst : aFirst) {
          sv[t] = vb; sv[ixj] = va; si[t] = ib; si[ixj] = ia;
        }
      }
      __syncthreads();
    }
  }

  float maxv = sv[1];                      // rank-1 after dropping top-1
  float e = 0.0f;
  if (t >= 1 && t <= 200) e = __expf((sv[t] - maxv) * INV_T);
  red[t] = e;
  for (int c = t; c < 4 * NCLS; c += 256) votes[c] = 0.0f;
  if (t >= 1 && t <= 200) lbl[t] = labels[si[t]];
  __syncthreads();

  for (int s = 128; s > 0; s >>= 1) {
    if (t < s) red[t] += red[t + s];
    __syncthreads();
  }
  float sum = red[0];
  wsh[t] = e / sum;
  __syncthreads();

  if (t == 0) {                            // serialized => bit-deterministic
    for (int j = 1; j <= 200; ++j) {
      float w = wsh[j]; int c = lbl[j]; int r = j - 1;
      if (r < 10)  votes[c]            += w;
      if (r < 20)  votes[NCLS + c]     += w;
      if (r < 100) votes[2 * NCLS + c] += w;
      votes[3 * NCLS + c] += w;            // r < 200 always
    }
  }
  __syncthreads();

  for (int idx = t; idx < 4 * NCLS; idx += 256) {
    int kk = idx / NCLS, c = idx % NCLS;
    out[(size_t)kk * B_Q * NCLS + (size_t)row * NCLS + c] = votes[idx];
  }
}

// ---------------------------------------------------------------------------
extern "C" void kernel_launch(void* const* d_in, const int* in_sizes, int n_in,
                              void* d_out, int out_size, void* d_ws, size_t ws_size,
                              hipStream_t stream)
{
  (void)in_sizes; (void)n_in; (void)out_size; (void)ws_size;
  const float* feat   = (const float*)d_in[0];   // (2048, 1024)
  const float* train  = (const float*)d_in[1];   // (100000, 1024)
  const int*   labels = (const int*)d_in[2];     // (100000,)
  float* out = (float*)d_out;                    // (4, 2048, 1000)

  char* ws = (char*)d_ws;
  size_t off = 0;
  float* sims = (float*)(ws + off); off += (size_t)B_Q * CHUNK   * sizeof(float);
  float* hv   = (float*)(ws + off); off += (size_t)B_Q * HSTRIDE * sizeof(float);
  int*   hi   = (int*)  (ws + off); off += (size_t)B_Q * HSTRIDE * sizeof(int);
  int*   hs   = (int*)  (ws + off); off += (size_t)B_Q * sizeof(int);

  dim3 ggrid(CHUNK / 128, B_Q / 64);   // 49 x 32 blocks, 64x128 tile per block
  for (int c = 0; c < NCHUNK; ++c) {
    int chunk_off = c * CHUNK;
    gemm_chunk_kernel<<<ggrid, 256, 0, stream>>>(feat, train, sims, chunk_off);
    topk_update_kernel<<<B_Q, 256, 0, stream>>>(sims, hv, hi, hs, c, chunk_off);
  }
  finalize_kernel<<<B_Q, 256, 0, stream>>>(hv, hi, labels, out);
}